// RGCNEncoder_60876866453608
// MI455X (gfx1250) — compile-verified
//
#include <hip/hip_runtime.h>

#define N_NODES 50000
#define NREL    16
#define DIM     200
#define NEDGE   400000

typedef __attribute__((ext_vector_type(2))) float v2f;
typedef __attribute__((ext_vector_type(8))) float v8f;

// ---------------------------------------------------------------------------
// small elementwise helpers
// ---------------------------------------------------------------------------
__global__ void zero_kernel(float* __restrict__ p, int n) {
    int i = blockIdx.x * blockDim.x + threadIdx.x;
    if (i < n) p[i] = 0.0f;
}

__global__ void count_kernel(const int* __restrict__ etype,
                             const int* __restrict__ dst,
                             float* __restrict__ cnt) {
    int e = blockIdx.x * blockDim.x + threadIdx.x;
    if (e < NEDGE) atomicAdd(&cnt[etype[e] * N_NODES + dst[e]], 1.0f);
}

__global__ void invert_kernel(float* __restrict__ cnt, int n) {
    int i = blockIdx.x * blockDim.x + threadIdx.x;
    if (i < n) {
        float c = cnt[i];
        cnt[i] = 1.0f / (c > 1.0f ? c : 1.0f);   // matches jnp.clip(cnt, 1.0)
    }
}

__global__ void relu_kernel(float* __restrict__ p, int n) {
    int i = blockIdx.x * blockDim.x + threadIdx.x;
    if (i < n) {
        float v = p[i];
        p[i] = v > 0.0f ? v : 0.0f;
    }
}

// ---------------------------------------------------------------------------
// C[50000 x 200] = A[50000 x 200] @ B[200 x 200] (+ bias broadcast per column)
// One wave32 computes a 32x32 C block = 2x2 tiles of 16x16 via
// V_WMMA_F32_16X16X4_F32 (2x2 register blocking: 6 loads feed 4 WMMAs/K-step).
// ---------------------------------------------------------------------------
#define ROW_TILES (N_NODES / 16)         // 3125
#define COL_TILES 13                     // covers 208 >= 200 columns
#define RBLK ((ROW_TILES + 1) / 2)       // 1563 row blocks of 32 rows
#define CBLK ((COL_TILES + 1) / 2)       // 7 col blocks of 32 cols

__launch_bounds__(256)
__global__ void gemm_wmma(const float* __restrict__ A,
                          const float* __restrict__ B,
                          const float* __restrict__ bias,
                          float* __restrict__ C) {
    int wave = (int)((blockIdx.x * blockDim.x + threadIdx.x) >> 5);
    if (wave >= RBLK * CBLK) return;

    int rb   = wave / CBLK;
    int cb   = wave % CBLK;
    int lane = threadIdx.x & 31;
    int hi   = lane >> 4;        // 0: lanes 0-15, 1: lanes 16-31
    int lo   = lane & 15;

    int r0 = rb * 32;
    int c0 = cb * 32;

    // A source rows for the two row tiles (clamped; junk only pollutes
    // unstored C rows of the final partial block)
    int row0 = r0 + lo;
    int row1 = r0 + 16 + lo;
    int rc0  = row0 < N_NODES ? row0 : N_NODES - 1;
    int rc1  = row1 < N_NODES ? row1 : N_NODES - 1;

    // B/C columns for the two col tiles (clamped; DIM % 16 == 8)
    int n0  = c0 + lo;
    int n1  = c0 + 16 + lo;
    int nc0 = n0 < DIM ? n0 : DIM - 1;
    int nc1 = n1 < DIM ? n1 : DIM - 1;

    // C/D layout: VGPR v holds rows (tileRow + v + 8*hi), column (lane&15).
    v8f acc00, acc01, acc10, acc11;
    float bi0 = bias ? bias[nc0] : 0.0f;
    float bi1 = bias ? bias[nc1] : 0.0f;
#pragma unroll
    for (int v = 0; v < 8; ++v) {
        acc00[v] = bi0; acc01[v] = bi1;
        acc10[v] = bi0; acc11[v] = bi1;
    }

    // A layout: lane (hi,lo) supplies A[row][k + 2*hi] (VGPR0) and +1 (VGPR1)
    const float* ap0 = A + (size_t)rc0 * DIM + 2 * hi;
    const float* ap1 = A + (size_t)rc1 * DIM + 2 * hi;
    // B layout: lane supplies B[k + 2*hi][n] (VGPR0) and B[k + 2*hi + 1][n] (VGPR1)
    const float* bp0 = B + (size_t)(2 * hi) * DIM + nc0;
    const float* bp1 = B + (size_t)(2 * hi) * DIM + nc1;

    for (int k = 0; k < DIM; k += 4) {
        v2f a0; a0[0] = ap0[0];   a0[1] = ap0[1];
        v2f a1; a1[0] = ap1[0];   a1[1] = ap1[1];
        v2f b0; b0[0] = bp0[0];   b0[1] = bp0[DIM];
        v2f b1; b1[0] = bp1[0];   b1[1] = bp1[DIM];

        acc00 = __builtin_amdgcn_wmma_f32_16x16x4_f32(false, a0, false, b0,
                                                      (short)0, acc00, false, false);
        acc01 = __builtin_amdgcn_wmma_f32_16x16x4_f32(false, a0, false, b1,
                                                      (short)0, acc01, false, false);
        acc10 = __builtin_amdgcn_wmma_f32_16x16x4_f32(false, a1, false, b0,
                                                      (short)0, acc10, false, false);
        acc11 = __builtin_amdgcn_wmma_f32_16x16x4_f32(false, a1, false, b1,
                                                      (short)0, acc11, false, false);

        ap0 += 4; ap1 += 4;
        bp0 += 4 * DIM; bp1 += 4 * DIM;
    }

    // Guarded tile stores. Row guard is tile-uniform (N_NODES % 16 == 0).
    {
        int tr, col;
        tr = r0;      col = n0;
        if (tr < N_NODES && col < DIM) {
            float* cp = C + (size_t)(tr + hi * 8) * DIM + col;
#pragma unroll
            for (int v = 0; v < 8; ++v) cp[(size_t)v * DIM] = acc00[v];
        }
        tr = r0;      col = n1;
        if (tr < N_NODES && col < DIM) {
            float* cp = C + (size_t)(tr + hi * 8) * DIM + col;
#pragma unroll
            for (int v = 0; v < 8; ++v) cp[(size_t)v * DIM] = acc01[v];
        }
        tr = r0 + 16; col = n0;
        if (tr < N_NODES && col < DIM) {
            float* cp = C + (size_t)(tr + hi * 8) * DIM + col;
#pragma unroll
            for (int v = 0; v < 8; ++v) cp[(size_t)v * DIM] = acc10[v];
        }
        tr = r0 + 16; col = n1;
        if (tr < N_NODES && col < DIM) {
            float* cp = C + (size_t)(tr + hi * 8) * DIM + col;
#pragma unroll
            for (int v = 0; v < 8; ++v) cp[(size_t)v * DIM] = acc11[v];
        }
    }
}

// ---------------------------------------------------------------------------
// Per-relation edge scatter: out[dst] += xw[src] * inv_cnt[rel, dst]
// One wave per edge; wave-uniform relation filter.
// ---------------------------------------------------------------------------
__launch_bounds__(256)
__global__ void scatter_kernel(const float* __restrict__ xw,
                               const int* __restrict__ src,
                               const int* __restrict__ dst,
                               const int* __restrict__ etype,
                               const float* __restrict__ inv,
                               float* __restrict__ out, int rel) {
    int wid = (int)((blockIdx.x * blockDim.x + threadIdx.x) >> 5);
    if (wid >= NEDGE) return;
    if (etype[wid] != rel) return;             // uniform across the wave

    int lane = threadIdx.x & 31;
    int s = src[wid];
    int d = dst[wid];
    float w = inv[rel * N_NODES + d];

    const float* xr   = xw  + (size_t)s * DIM;
    float*       orow = out + (size_t)d * DIM;
    for (int c = lane; c < DIM; c += 32)
        atomicAdd(&orow[c], xr[c] * w);
}

// ---------------------------------------------------------------------------
// launch
// ---------------------------------------------------------------------------
extern "C" void kernel_launch(void* const* d_in, const int* in_sizes, int n_in,
                              void* d_out, int out_size, void* d_ws, size_t ws_size,
                              hipStream_t stream) {
    const int*   edge_index = (const int*)d_in[0];   // [2, E]
    const int*   etype      = (const int*)d_in[1];   // [E]
    const float* emb        = (const float*)d_in[2]; // [N, D]
    const float* W1         = (const float*)d_in[3]; // [R, D, D]
    const float* root1      = (const float*)d_in[4]; // [D, D]
    const float* b1         = (const float*)d_in[5]; // [D]
    const float* W2         = (const float*)d_in[6];
    const float* root2      = (const float*)d_in[7];
    const float* b2         = (const float*)d_in[8];

    const int* src = edge_index;
    const int* dst = edge_index + NEDGE;
    float* out = (float*)d_out;

    // workspace layout (fp32): counts | xw | h1   (~83 MB)
    float* cnt = (float*)d_ws;                             // R*N
    float* xw  = cnt + (size_t)NREL * N_NODES;             // N*D
    float* h1  = xw + (size_t)N_NODES * DIM;               // N*D

    const int RN = NREL * N_NODES;

    // --- per-(relation,dst) mean denominators (graph shared by both layers) ---
    zero_kernel<<<(RN + 255) / 256, 256, 0, stream>>>(cnt, RN);
    count_kernel<<<(NEDGE + 255) / 256, 256, 0, stream>>>(etype, dst, cnt);
    invert_kernel<<<(RN + 255) / 256, 256, 0, stream>>>(cnt, RN);

    const int GWAVES = RBLK * CBLK;                        // 10941 waves
    const int ggrid  = (GWAVES * 32 + 255) / 256;
    const int sgrid  = (int)(((long long)NEDGE * 32 + 255) / 256); // 50000 blocks
    const int nelem  = N_NODES * DIM;

    // ---------------- layer 1 ----------------
    gemm_wmma<<<ggrid, 256, 0, stream>>>(emb, root1, b1, h1);   // h1 = emb@root1 + b1
    for (int r = 0; r < NREL; ++r) {
        gemm_wmma<<<ggrid, 256, 0, stream>>>(emb, W1 + (size_t)r * DIM * DIM, nullptr, xw);
        scatter_kernel<<<sgrid, 256, 0, stream>>>(xw, src, dst, etype, cnt, h1, r);
    }
    relu_kernel<<<(nelem + 255) / 256, 256, 0, stream>>>(h1, nelem);

    // ---------------- layer 2 ----------------
    gemm_wmma<<<ggrid, 256, 0, stream>>>(h1, root2, b2, out);   // out = h1@root2 + b2
    for (int r = 0; r < NREL; ++r) {
        gemm_wmma<<<ggrid, 256, 0, stream>>>(h1, W2 + (size_t)r * DIM * DIM, nullptr, xw);
        scatter_kernel<<<sgrid, 256, 0, stream>>>(xw, src, dst, etype, cnt, out, r);
    }
}